// GNN_56762287784201
// MI455X (gfx1250) — compile-verified
//
#include <hip/hip_runtime.h>
#include <hip/hip_bf16.h>

typedef __attribute__((ext_vector_type(2))) float v2f;
typedef __attribute__((ext_vector_type(8))) float v8f;

// ---------------------------------------------------------------------------
// Zero-fill (grid-stride). Needed every call: scatter kernels accumulate.
// ---------------------------------------------------------------------------
__global__ void gnn_zero_kernel(float* __restrict__ p, long long n) {
    long long t = (long long)blockIdx.x * blockDim.x + threadIdx.x;
    long long stride = (long long)gridDim.x * blockDim.x;
    for (long long i = t; i < n; i += stride) p[i] = 0.0f;
}

// ---------------------------------------------------------------------------
// In-degree counts: cnt[dst[e]] += 1.0  (one thread per edge)
// ---------------------------------------------------------------------------
__global__ void gnn_count_kernel(const int* __restrict__ dst, float* __restrict__ cnt, int E) {
    int e = blockIdx.x * blockDim.x + threadIdx.x;
    if (e < E) atomicAdd(&cnt[dst[e]], 1.0f);
}

// ---------------------------------------------------------------------------
// Scatter-add features: agg[dst[e], :] += feat[src[e], :]
// One wave32 per edge; each lane owns a float4 (4*32 = 128 columns).
// src/dst are wave-uniform -> scalarized loads; x gather hits L2 (51MB < 192MB).
// ---------------------------------------------------------------------------
__global__ void gnn_scatter_kernel(const float* __restrict__ feat,
                                   const int* __restrict__ src,
                                   const int* __restrict__ dst,
                                   float* __restrict__ agg, int E) {
    long long t = (long long)blockIdx.x * blockDim.x + threadIdx.x;
    int e = (int)(t >> 5);
    if (e >= E) return;
    int lane = (int)(t & 31);
    int s = src[e];
    int d = dst[e];
    const float4 v = ((const float4*)(feat + (long long)s * 128))[lane];
    float* base = agg + (long long)d * 128 + lane * 4;
    atomicAdd(base + 0, v.x);
    atomicAdd(base + 1, v.y);
    atomicAdd(base + 2, v.z);
    atomicAdd(base + 3, v.w);
}

// ---------------------------------------------------------------------------
// Fused SAGE dense stage:
//   out = [relu]( (agg/max(cnt,1)) @ wl + xin @ wr + bias )
// Block = 256 threads = 8 waves; block -> one 16-row tile, wave w -> cols
// [16w, 16w+16). K-loop in steps of 4 using V_WMMA_F32_16X16X4_F32.
//
// A frag (16x4 f32): lanes 0-15 hold K={0,1}, lanes 16-31 hold K={2,3}; M=lane&15.
// B frag (4x16 f32): lanes 0-15 hold K={0,1}, lanes 16-31 hold K={2,3}; N=lane&15.
// C/D (16x16 f32):   VGPR r -> M = r + 8*lane[4], N = lane&15.
// ---------------------------------------------------------------------------
__global__ __launch_bounds__(256)
void gnn_sage_gemm_kernel(const float* __restrict__ agg,
                          const float* __restrict__ cnt,
                          const float* __restrict__ xin,
                          const float* __restrict__ wl,
                          const float* __restrict__ wr,
                          const float* __restrict__ bias,
                          float* __restrict__ out,
                          int do_relu) {
    const int lane    = threadIdx.x & 31;
    const int wave    = threadIdx.x >> 5;      // column tile 0..7
    const int rowbase = blockIdx.x * 16;
    const int m       = lane & 15;             // A row within tile / D col
    const int hi      = lane >> 4;             // 0 or 1
    const int khalf   = hi * 2;                // K sub-offset {0,2}
    const int col     = wave * 16 + m;         // output column

    const float c     = cnt[rowbase + m];
    const float scale = 1.0f / fmaxf(c, 1.0f);

    const float* arow = agg + (long long)(rowbase + m) * 128 + khalf;
    const float* xrow = xin + (long long)(rowbase + m) * 128 + khalf;

    v8f acc = {};

    // mean @ wl  (mean division fused into the A fragment)
#pragma unroll
    for (int k0 = 0; k0 < 128; k0 += 4) {
        v2f a = *(const v2f*)(arow + k0);
        a = a * scale;
        v2f b;
        b.x = wl[(long long)(k0 + khalf) * 128 + col];
        b.y = wl[(long long)(k0 + khalf + 1) * 128 + col];
        acc = __builtin_amdgcn_wmma_f32_16x16x4_f32(false, a, false, b,
                                                    (short)0, acc, false, false);
    }
    // xin @ wr
#pragma unroll
    for (int k0 = 0; k0 < 128; k0 += 4) {
        v2f a = *(const v2f*)(xrow + k0);
        v2f b;
        b.x = wr[(long long)(k0 + khalf) * 128 + col];
        b.y = wr[(long long)(k0 + khalf + 1) * 128 + col];
        acc = __builtin_amdgcn_wmma_f32_16x16x4_f32(false, a, false, b,
                                                    (short)0, acc, false, false);
    }

    const float bv = bias[col];
    const int rowoff = hi * 8;
#pragma unroll
    for (int r = 0; r < 8; ++r) {
        float v = acc[r] + bv;
        if (do_relu) v = fmaxf(v, 0.0f);
        out[(long long)(rowbase + rowoff + r) * 128 + col] = v;
    }
}

// ---------------------------------------------------------------------------
// Host-side orchestration
// ---------------------------------------------------------------------------
extern "C" void kernel_launch(void* const* d_in, const int* in_sizes, int n_in,
                              void* d_out, int out_size, void* d_ws, size_t ws_size,
                              hipStream_t stream) {
    const int D = 128;
    const int N = in_sizes[0] / D;       // 100000
    const int E = in_sizes[1] / 2;       // 1600000

    const float* x    = (const float*)d_in[0];
    const int*   adj  = (const int*)d_in[1];
    const float* w_l0 = (const float*)d_in[2];
    const float* w_r0 = (const float*)d_in[3];
    const float* b0   = (const float*)d_in[4];
    const float* w_l1 = (const float*)d_in[5];
    const float* w_r1 = (const float*)d_in[6];
    const float* b1   = (const float*)d_in[7];
    float*       out  = (float*)d_out;

    const int* src = adj;
    const int* dst = adj + E;

    // Workspace layout (floats): cnt[N] | agg[N*D] (reused by both layers) | h[N*D]
    float* cnt = (float*)d_ws;
    float* agg = cnt + N;
    float* h   = agg + (long long)N * D;

    const int T = 256;
    const long long scatter_threads = (long long)E * 32;
    const int scatter_blocks = (int)((scatter_threads + T - 1) / T);
    const int edge_blocks    = (E + T - 1) / T;
    const int gemm_blocks    = N / 16;   // N divisible by 16 -> EXEC all-1s for WMMA
    const int zero_blocks    = 4096;

    // ---- Layer 0 ----
    // zero cnt + agg in one pass (they are adjacent)
    gnn_zero_kernel<<<zero_blocks, T, 0, stream>>>(cnt, (long long)N * (D + 1));
    gnn_count_kernel<<<edge_blocks, T, 0, stream>>>(dst, cnt, E);
    gnn_scatter_kernel<<<scatter_blocks, T, 0, stream>>>(x, src, dst, agg, E);
    gnn_sage_gemm_kernel<<<gemm_blocks, T, 0, stream>>>(agg, cnt, x, w_l0, w_r0, b0, h, 1);

    // ---- Layer 1 ----
    gnn_zero_kernel<<<zero_blocks, T, 0, stream>>>(agg, (long long)N * D);
    gnn_scatter_kernel<<<scatter_blocks, T, 0, stream>>>(h, src, dst, agg, E);
    gnn_sage_gemm_kernel<<<gemm_blocks, T, 0, stream>>>(agg, cnt, h, w_l1, w_r1, b1, out, 0);
}